// offset_deform_9165460210005
// MI455X (gfx1250) — compile-verified
//
#include <hip/hip_runtime.h>
#include <hip/hip_bf16.h>
#include <cstdint>
#include <cstddef>

#define BB   4
#define NN   2048
#define KNB  30
#define MN   (BB*NN)          // 8192 rows (B*N), multiple of 64
#define MEK  (BB*NN*KNB)      // 245760 rows (B*N*K), multiple of 64
#define EPSB 1e-5f

typedef float v2f __attribute__((ext_vector_type(2)));
typedef float v8f __attribute__((ext_vector_type(8)));

static __device__ __forceinline__ v8f wmma4(v2f a, v2f b, v8f c) {
  // D(16x16,f32) = A(16x4,f32) * B(4x16,f32) + C
  return __builtin_amdgcn_wmma_f32_16x16x4_f32(
      false, a, false, b, (short)0, c, false, false);
}
static __device__ __forceinline__ v8f vzero8() {
  v8f z = {0.f, 0.f, 0.f, 0.f, 0.f, 0.f, 0.f, 0.f};
  return z;
}
static __device__ __forceinline__ int imin(int a, int b) { return a < b ? a : b; }

// ------------------------------------------------------------- transpose in
__global__ void k_transpose(const float* __restrict__ in, float* __restrict__ pc) {
  int i = blockIdx.x * blockDim.x + threadIdx.x;
  if (i >= MN * 3) return;
  int b = i / (NN * 3);
  int r = i - b * (NN * 3);
  int n = r / 3;
  int c = r - n * 3;
  pc[i] = in[(b * 3 + c) * NN + n];
}

// -------------------------------------------------------- KNN on 3-channel
__global__ void k_knn3(const float* __restrict__ pts, int* __restrict__ idx) {
  __shared__ float sx[256], sy[256], sz[256];
  int tid = threadIdx.x;
  int b = blockIdx.y;
  int n = blockIdx.x * 256 + tid;
  const float* p = pts + (size_t)(b * NN + n) * 3;
  float px = p[0], py = p[1], pz = p[2];
  float bd[KNB];
  int   bi[KNB];
#pragma unroll
  for (int i = 0; i < KNB; ++i) { bd[i] = 3.4e38f; bi[i] = 0; }
  float worst = 3.4e38f;                     // register copy of bd[KNB-1]
  for (int t = 0; t < NN; t += 256) {
    __syncthreads();
    const float* q = pts + (size_t)(b * NN + t + tid) * 3;
    sx[tid] = q[0]; sy[tid] = q[1]; sz[tid] = q[2];
    __syncthreads();
    for (int j = 0; j < 256; ++j) {
      float dx = px - sx[j], dy = py - sy[j], dz = pz - sz[j];
      float d = dx * dx + dy * dy + dz * dz;
      if (d < worst) {
        int pos = KNB - 1;
        while (pos > 0 && bd[pos - 1] > d) {
          bd[pos] = bd[pos - 1]; bi[pos] = bi[pos - 1]; --pos;
        }
        bd[pos] = d; bi[pos] = t + j;
        worst = bd[KNB - 1];
      }
    }
  }
  int* o = idx + (size_t)(b * NN + n) * KNB;
#pragma unroll
  for (int i = 0; i < KNB; ++i) o[i] = bi[i];
}

// -------------------------------------------- plain GEMM: Y = X @ W^T (WMMA)
// X: (M, Cin) row-major, W: (Cout, Cin) row-major, Y: (M, Cout) row-major.
// One wave computes a 64x16 output slab (4 M-tiles sharing one B fragment).
// Ping-pong double buffering: loads for a k-group issue one group ahead of
// the WMMAs that consume it; no register rotation copies. M % 64 == 0.
__global__ void k_gemm(const float* __restrict__ A, const float* __restrict__ W,
                       float* __restrict__ Y, int M, int Cin, int Cout) {
  int wave = (blockIdx.x * blockDim.x + threadIdx.x) >> 5;
  int lane = threadIdx.x & 31;
  int ntN = (Cout + 15) >> 4;
  int ntG = M >> 6;
  if (wave >= ntG * ntN) return;              // wave-uniform branch
  int tg = wave / ntN, tn = wave - tg * ntN;
  int half = lane >> 4, r = lane & 15;
  int col = tn * 16 + r;
  bool colv = (col < Cout);
  const float* Wr = W + (size_t)(colv ? col : (Cout - 1)) * Cin;
  const float* Ar = A + (size_t)(tg * 64 + r) * Cin;
  const size_t As = (size_t)16 * Cin;
  v8f c0 = vzero8(), c1 = vzero8(), c2 = vzero8(), c3 = vzero8();

#define GLOAD(BF, F0, F1, F2, F3, KK) do {                       \
    int k0_ = (KK) + half * 2;                                   \
    BF.x = Wr[k0_];            BF.y = Wr[k0_ + 1];               \
    F0.x = Ar[k0_];            F0.y = Ar[k0_ + 1];               \
    F1.x = Ar[As + k0_];       F1.y = Ar[As + k0_ + 1];          \
    F2.x = Ar[2 * As + k0_];   F2.y = Ar[2 * As + k0_ + 1];      \
    F3.x = Ar[3 * As + k0_];   F3.y = Ar[3 * As + k0_ + 1];      \
  } while (0)
#define GMMA(BF, F0, F1, F2, F3) do {                            \
    c0 = wmma4(F0, BF, c0);                                      \
    c1 = wmma4(F1, BF, c1);                                      \
    c2 = wmma4(F2, BF, c2);                                      \
    c3 = wmma4(F3, BF, c3);                                      \
  } while (0)

  v2f bA, aA0, aA1, aA2, aA3;                 // buffer A
  v2f bX, aX0, aX1, aX2, aX3;                 // buffer B
  GLOAD(bA, aA0, aA1, aA2, aA3, 0);
  int kb = 4;
#pragma clang loop unroll(disable)
  for (; kb + 7 < Cin; kb += 8) {
    GLOAD(bX, aX0, aX1, aX2, aX3, kb);        // group g+1 in flight
    GMMA(bA, aA0, aA1, aA2, aA3);             // consume group g
    GLOAD(bA, aA0, aA1, aA2, aA3, kb + 4);    // group g+2 in flight
    GMMA(bX, aX0, aX1, aX2, aX3);             // consume group g+1
  }
  if (kb + 3 < Cin) {                         // one leftover full group
    GLOAD(bX, aX0, aX1, aX2, aX3, kb);
    GMMA(bA, aA0, aA1, aA2, aA3);
    GMMA(bX, aX0, aX1, aX2, aX3);
    kb += 4;
  } else {
    GMMA(bA, aA0, aA1, aA2, aA3);
  }
#undef GLOAD
#undef GMMA

  if (kb < Cin) {                             // guarded partial tail
    int k0 = kb + half * 2;
    int q0 = imin(k0, Cin - 1), q1 = imin(k0 + 1, Cin - 1);
    bool v0 = (k0 < Cin), v1 = (k0 + 1 < Cin);
    v2f tb, t0, t1, t2, t3;
    tb.x = v0 ? Wr[q0] : 0.f;            tb.y = v1 ? Wr[q1] : 0.f;
    t0.x = v0 ? Ar[q0] : 0.f;            t0.y = v1 ? Ar[q1] : 0.f;
    t1.x = v0 ? Ar[As + q0] : 0.f;       t1.y = v1 ? Ar[As + q1] : 0.f;
    t2.x = v0 ? Ar[2 * As + q0] : 0.f;   t2.y = v1 ? Ar[2 * As + q1] : 0.f;
    t3.x = v0 ? Ar[3 * As + q0] : 0.f;   t3.y = v1 ? Ar[3 * As + q1] : 0.f;
    c0 = wmma4(t0, tb, c0);
    c1 = wmma4(t1, tb, c1);
    c2 = wmma4(t2, tb, c2);
    c3 = wmma4(t3, tb, c3);
  }

  if (colv) {
    size_t mb = (size_t)(tg * 64 + half * 8);
#pragma unroll
    for (int i = 0; i < 8; ++i) Y[(mb + i) * Cout + col]      = c0[i];
#pragma unroll
    for (int i = 0; i < 8; ++i) Y[(mb + 16 + i) * Cout + col] = c1[i];
#pragma unroll
    for (int i = 0; i < 8; ++i) Y[(mb + 32 + i) * Cout + col] = c2[i];
#pragma unroll
    for (int i = 0; i < 8; ++i) Y[(mb + 48 + i) * Cout + col] = c3[i];
  }
}

// ---------------- fused edge-feature + GEMM: Y = edgefeat(X, idx) @ W^T
// X: (B, N, Cpt); edge row (b,n,k): [X[b,n,:], X[b,idx,:]-X[b,n,:]] (2*Cpt)
// One wave computes 64x16 outputs. M = MEK (multiple of 64).
__device__ __forceinline__ float edge_val(const float* Xs, const float* Xj,
                                          int k, int Cpt) {
  bool self = (k < Cpt);
  int cc = self ? k : (k - Cpt);
  float xs = Xs[cc];
  float xj = Xj[cc];
  return self ? xs : (xj - xs);
}

__global__ void k_edge_gemm(const float* __restrict__ X, const int* __restrict__ idx,
                            const float* __restrict__ W, float* __restrict__ Y,
                            int Cpt, int Cout) {
  const int Cin = 2 * Cpt;
  int wave = (blockIdx.x * blockDim.x + threadIdx.x) >> 5;
  int lane = threadIdx.x & 31;
  int ntN = (Cout + 15) >> 4;
  int ntG = MEK >> 6;
  if (wave >= ntG * ntN) return;
  int tg = wave / ntN, tn = wave - tg * ntN;
  int half = lane >> 4, r = lane & 15;
  int col = tn * 16 + r;
  bool colv = (col < Cout);
  const float* Wr = W + (size_t)(colv ? col : (Cout - 1)) * Cin;

  // per-lane row decode for the 4 M-tiles
  const float* Xs[4];
  const float* Xj[4];
#pragma unroll
  for (int t = 0; t < 4; ++t) {
    int row = tg * 64 + t * 16 + r;
    int b  = row / (NN * KNB);
    int r2 = row - b * (NN * KNB);
    int n  = r2 / KNB;
    int kk = r2 - n * KNB;
    int j  = idx[(size_t)(b * NN + n) * KNB + kk];
    Xs[t] = X + (size_t)(b * NN + n) * Cpt;
    Xj[t] = X + (size_t)(b * NN + j) * Cpt;
  }

  v8f c0 = vzero8(), c1 = vzero8(), c2 = vzero8(), c3 = vzero8();

#define ELOAD(BF, F0, F1, F2, F3, KK) do {                                  \
    int k0_ = (KK) + half * 2;                                              \
    BF.x = Wr[k0_]; BF.y = Wr[k0_ + 1];                                     \
    F0.x = edge_val(Xs[0], Xj[0], k0_, Cpt);                                \
    F0.y = edge_val(Xs[0], Xj[0], k0_ + 1, Cpt);                            \
    F1.x = edge_val(Xs[1], Xj[1], k0_, Cpt);                                \
    F1.y = edge_val(Xs[1], Xj[1], k0_ + 1, Cpt);                            \
    F2.x = edge_val(Xs[2], Xj[2], k0_, Cpt);                                \
    F2.y = edge_val(Xs[2], Xj[2], k0_ + 1, Cpt);                            \
    F3.x = edge_val(Xs[3], Xj[3], k0_, Cpt);                                \
    F3.y = edge_val(Xs[3], Xj[3], k0_ + 1, Cpt);                            \
  } while (0)
#define EMMA(BF, F0, F1, F2, F3) do {                                       \
    c0 = wmma4(F0, BF, c0);                                                 \
    c1 = wmma4(F1, BF, c1);                                                 \
    c2 = wmma4(F2, BF, c2);                                                 \
    c3 = wmma4(F3, BF, c3);                                                 \
  } while (0)

  v2f bA, aA0, aA1, aA2, aA3;
  v2f bX, aX0, aX1, aX2, aX3;
  ELOAD(bA, aA0, aA1, aA2, aA3, 0);
  int kb = 4;
#pragma clang loop unroll(disable)
  for (; kb + 7 < Cin; kb += 8) {
    ELOAD(bX, aX0, aX1, aX2, aX3, kb);
    EMMA(bA, aA0, aA1, aA2, aA3);
    ELOAD(bA, aA0, aA1, aA2, aA3, kb + 4);
    EMMA(bX, aX0, aX1, aX2, aX3);
  }
  if (kb + 3 < Cin) {
    ELOAD(bX, aX0, aX1, aX2, aX3, kb);
    EMMA(bA, aA0, aA1, aA2, aA3);
    EMMA(bX, aX0, aX1, aX2, aX3);
    kb += 4;
  } else {
    EMMA(bA, aA0, aA1, aA2, aA3);
  }
#undef ELOAD
#undef EMMA

  if (kb < Cin) {                              // tail: clamp + select, no branchy loads
    int k0 = kb + half * 2;
    int q0 = imin(k0, Cin - 1), q1 = imin(k0 + 1, Cin - 1);
    bool v0 = (k0 < Cin), v1 = (k0 + 1 < Cin);
    v2f tb, t0, t1, t2, t3;
    tb.x = v0 ? Wr[q0] : 0.f; tb.y = v1 ? Wr[q1] : 0.f;
    t0.x = v0 ? edge_val(Xs[0], Xj[0], q0, Cpt) : 0.f;
    t0.y = v1 ? edge_val(Xs[0], Xj[0], q1, Cpt) : 0.f;
    t1.x = v0 ? edge_val(Xs[1], Xj[1], q0, Cpt) : 0.f;
    t1.y = v1 ? edge_val(Xs[1], Xj[1], q1, Cpt) : 0.f;
    t2.x = v0 ? edge_val(Xs[2], Xj[2], q0, Cpt) : 0.f;
    t2.y = v1 ? edge_val(Xs[2], Xj[2], q1, Cpt) : 0.f;
    t3.x = v0 ? edge_val(Xs[3], Xj[3], q0, Cpt) : 0.f;
    t3.y = v1 ? edge_val(Xs[3], Xj[3], q1, Cpt) : 0.f;
    c0 = wmma4(t0, tb, c0);
    c1 = wmma4(t1, tb, c1);
    c2 = wmma4(t2, tb, c2);
    c3 = wmma4(t3, tb, c3);
  }

  if (colv) {
    size_t mb = (size_t)(tg * 64 + half * 8);
#pragma unroll
    for (int i = 0; i < 8; ++i) Y[(mb + i) * Cout + col]      = c0[i];
#pragma unroll
    for (int i = 0; i < 8; ++i) Y[(mb + 16 + i) * Cout + col] = c1[i];
#pragma unroll
    for (int i = 0; i < 8; ++i) Y[(mb + 32 + i) * Cout + col] = c2[i];
#pragma unroll
    for (int i = 0; i < 8; ++i) Y[(mb + 48 + i) * Cout + col] = c3[i];
  }
}

// ---------------------------------------------------- BatchNorm statistics
__global__ void k_bn_stats(const float* __restrict__ Y, int M, int C,
                           float* __restrict__ sum, float* __restrict__ sumsq) {
  __shared__ float s1[256], s2[256];
  int c = blockIdx.x;
  float a = 0.f, a2 = 0.f;
  for (int i = threadIdx.x; i < M; i += 256) {
    float v = Y[(size_t)i * C + c];
    a += v; a2 += v * v;
  }
  s1[threadIdx.x] = a; s2[threadIdx.x] = a2;
  __syncthreads();
  for (int s = 128; s > 0; s >>= 1) {
    if (threadIdx.x < s) {
      s1[threadIdx.x] += s1[threadIdx.x + s];
      s2[threadIdx.x] += s2[threadIdx.x + s];
    }
    __syncthreads();
  }
  if (threadIdx.x == 0) { sum[c] = s1[0]; sumsq[c] = s2[0]; }
}

// ----------------------------------------- BN + LeakyReLU apply (in-place)
__global__ void k_bn_apply(float* __restrict__ Y, int M, int C,
                           const float* __restrict__ sum, const float* __restrict__ sumsq,
                           const float* __restrict__ g, const float* __restrict__ bta) {
  size_t total = (size_t)M * C;
  size_t stride = (size_t)gridDim.x * blockDim.x;
  float invM = 1.0f / (float)M;
  for (size_t i = (size_t)blockIdx.x * blockDim.x + threadIdx.x; i < total; i += stride) {
    int c = (int)(i % C);
    float mean = sum[c] * invM;
    float var = sumsq[c] * invM - mean * mean;
    float y = (Y[i] - mean) * rsqrtf(var + EPSB) * g[c] + bta[c];
    Y[i] = y > 0.f ? y : 0.2f * y;
  }
}

// ---------------------------- max / mean over K into strided concat output
__global__ void k_kreduce(const float* __restrict__ Y, float* __restrict__ out,
                          int C, int ostride, int maxbase, int meanbase, int domean) {
  int i = blockIdx.x * blockDim.x + threadIdx.x;
  if (i >= MN * C) return;
  int row = i / C, c = i - row * C;
  const float* p = Y + ((size_t)row * KNB) * C + c;
  float mx = -3.4e38f, s = 0.f;
  for (int k = 0; k < KNB; ++k) {
    float v = p[(size_t)k * C];
    mx = v > mx ? v : mx;
    s += v;
  }
  out[(size_t)row * ostride + maxbase + c] = mx;
  if (domean) out[(size_t)row * ostride + meanbase + c] = s * (1.0f / KNB);
}

// -------------------------------------------------------------- max over N
__global__ void k_nmax(const float* __restrict__ Y, float* __restrict__ out, int C) {
  int i = blockIdx.x * blockDim.x + threadIdx.x;
  if (i >= BB * C) return;
  int b = i / C, c = i - b * C;
  float mx = -3.4e38f;
  for (int n = 0; n < NN; ++n) {
    float v = Y[(size_t)(b * NN + n) * C + c];
    mx = v > mx ? v : mx;
  }
  out[i] = mx;
}

// -------------------------------------------------------------- tiny FC
__global__ void k_fc(const float* __restrict__ X, const float* __restrict__ W,
                     const float* __restrict__ bias, float* __restrict__ Y,
                     int M, int Cin, int Cout) {
  int i = blockIdx.x * blockDim.x + threadIdx.x;
  if (i >= M * Cout) return;
  int m = i / Cout, o = i - m * Cout;
  float s = bias[o];
  for (int k = 0; k < Cin; ++k) s += X[(size_t)m * Cin + k] * W[(size_t)o * Cin + k];
  Y[i] = s;
}

// ----------------------------------------- transform = net@tw + eye + tb
__global__ void k_tbuild(const float* __restrict__ net, const float* __restrict__ tw,
                         const float* __restrict__ tb, float* __restrict__ T) {
  int i = blockIdx.x * blockDim.x + threadIdx.x;
  if (i >= BB * 9) return;
  int b = i / 9, j = i - b * 9;
  float s = tb[j] + ((j == 0 || j == 4 || j == 8) ? 1.0f : 0.0f);
  for (int k = 0; k < 256; ++k) s += net[(size_t)b * 256 + k] * tw[(size_t)k * 9 + j];
  T[i] = s;
}

// ------------------------------------------------------ pct = pc @ T(b)
__global__ void k_tapply(const float* __restrict__ pc, const float* __restrict__ T,
                         float* __restrict__ pct) {
  int i = blockIdx.x * blockDim.x + threadIdx.x;
  if (i >= MN) return;
  int b = i / NN;
  float x = pc[(size_t)i * 3 + 0], y = pc[(size_t)i * 3 + 1], z = pc[(size_t)i * 3 + 2];
  const float* t = T + b * 9;
#pragma unroll
  for (int j = 0; j < 3; ++j)
    pct[(size_t)i * 3 + j] = x * t[j] + y * t[3 + j] + z * t[6 + j];
}

// --------------------------------------------------- IF = [pct, out3]
__global__ void k_concat_if(const float* __restrict__ pct, const float* __restrict__ out3,
                            float* __restrict__ IF) {
  int i = blockIdx.x * blockDim.x + threadIdx.x;
  if (i >= MN * 67) return;
  int row = i / 67, c = i - row * 67;
  IF[i] = (c < 3) ? pct[(size_t)row * 3 + c] : out3[(size_t)row * 64 + (c - 3)];
}

// ---- final pack: loop iterations are identical -> replicate 4x per graph
__global__ void k_pack(const float* __restrict__ out5, const float* __restrict__ o5,
                       float* __restrict__ outp) {
  int i = blockIdx.x * blockDim.x + threadIdx.x;
  if (i >= MN * 16) return;
  int c = i & 15;
  int row = i >> 4;
  int b = row / NN, n = row - b * NN;
  float v5 = out5[(size_t)row * 16 + c];
  float vm = o5[(size_t)row * 32 + c];
  float ve = o5[(size_t)row * 32 + 16 + c];
  const size_t sec = (size_t)BB * 64 * NN;
#pragma unroll
  for (int g = 0; g < 4; ++g) {
    size_t o = ((size_t)(b * 64 + g * 16 + c)) * NN + n;
    outp[o] = v5;
    outp[sec + o] = vm;
    outp[2 * sec + o] = ve;
  }
}

// =========================================================== host launcher
static inline int gblk(long long total) { return (int)((total + 255) / 256); }

extern "C" void kernel_launch(void* const* d_in, const int* in_sizes, int n_in,
                              void* d_out, int out_size, void* d_ws, size_t ws_size,
                              hipStream_t stream) {
  (void)in_sizes; (void)n_in; (void)out_size; (void)ws_size;
  const float* input_image = (const float*)d_in[0];
  const float* w64  = (const float*)d_in[1];
  const float* g64  = (const float*)d_in[2];
  const float* b64  = (const float*)d_in[3];
  const float* w128 = (const float*)d_in[4];
  const float* g128 = (const float*)d_in[5];
  const float* b128 = (const float*)d_in[6];
  const float* w1024 = (const float*)d_in[7];
  const float* g1024 = (const float*)d_in[8];
  const float* b1024 = (const float*)d_in[9];
  const float* wo1 = (const float*)d_in[10];
  const float* go1 = (const float*)d_in[11];
  const float* bo1 = (const float*)d_in[12];
  const float* wo2 = (const float*)d_in[13];
  const float* go2 = (const float*)d_in[14];
  const float* bo2 = (const float*)d_in[15];
  const float* wo3 = (const float*)d_in[16];
  const float* go3 = (const float*)d_in[17];
  const float* bo3 = (const float*)d_in[18];
  const float* w1 = (const float*)d_in[19];
  const float* g1 = (const float*)d_in[20];
  const float* b1 = (const float*)d_in[21];
  const float* w2 = (const float*)d_in[22];
  const float* g2 = (const float*)d_in[23];
  const float* b2 = (const float*)d_in[24];
  const float* w3 = (const float*)d_in[25];
  const float* g3 = (const float*)d_in[26];
  const float* b3 = (const float*)d_in[27];
  const float* fc1_w = (const float*)d_in[28];
  const float* fc1_b = (const float*)d_in[29];
  const float* fc2_w = (const float*)d_in[30];
  const float* fc2_b = (const float*)d_in[31];
  const float* tw = (const float*)d_in[32];
  const float* tb = (const float*)d_in[33];
  float* out = (float*)d_out;

  // ---------------- workspace layout
  char* base = (char*)d_ws;
  size_t off = 0;
  auto alloc = [&](size_t bytes) -> char* {
    char* p = base + off;
    off += (bytes + 255) & ~(size_t)255;
    return p;
  };
  float* pc     = (float*)alloc((size_t)MN * 3 * 4);
  float* pct    = (float*)alloc((size_t)MN * 3 * 4);
  float* trans1 = (float*)alloc((size_t)MN * 3 * 4);
  int*   idx    = (int*)  alloc((size_t)MEK * 4);
  float* sum    = (float*)alloc(1024 * 4);
  float* sumsq  = (float*)alloc(1024 * 4);
  float* gfeat  = (float*)alloc((size_t)BB * 1024 * 4);
  float* fc1o   = (float*)alloc((size_t)BB * 512 * 4);
  float* fc2o   = (float*)alloc((size_t)BB * 256 * 4);
  float* T      = (float*)alloc((size_t)BB * 9 * 4);
  float* IF     = (float*)alloc((size_t)MN * 67 * 4);
  float* cat1   = (float*)alloc((size_t)MN * 128 * 4);
  float* out3   = (float*)alloc((size_t)MN * 64 * 4);
  float* nmax128= (float*)alloc((size_t)MN * 128 * 4);
  float* o5     = (float*)alloc((size_t)MN * 32 * 4);
  float* out5   = (float*)alloc((size_t)MN * 16 * 4);
  float* bigB   = (float*)alloc((size_t)MEK * 64 * 4);   // y64 / y64o1 / y16
  float* bigA   = (float*)alloc((size_t)MEK * 128 * 4);  // y128 / y1024 / y64o2

  auto gemm = [&](const float* A, const float* W, float* Y, int M, int Cin, int Cout) {
    int nwaves = (M >> 6) * ((Cout + 15) / 16);
    k_gemm<<<dim3((nwaves + 3) / 4), dim3(128), 0, stream>>>(A, W, Y, M, Cin, Cout);
  };
  auto edge_gemm = [&](const float* X, const int* I, const float* W, float* Y,
                       int Cpt, int Cout) {
    int nwaves = (MEK >> 6) * ((Cout + 15) / 16);
    k_edge_gemm<<<dim3((nwaves + 3) / 4), dim3(128), 0, stream>>>(X, I, W, Y, Cpt, Cout);
  };
  auto bn = [&](float* Y, int M, int C, const float* g, const float* bt) {
    k_bn_stats<<<dim3(C), dim3(256), 0, stream>>>(Y, M, C, sum, sumsq);
    long long tot = (long long)M * C;
    int blocks = (int)((tot + 255) / 256);
    if (blocks > 4096) blocks = 4096;
    k_bn_apply<<<dim3(blocks), dim3(256), 0, stream>>>(Y, M, C, sum, sumsq, g, bt);
  };

  // ---------------- stage 0: transpose + knn on pc
  k_transpose<<<dim3(gblk((long long)MN * 3)), dim3(256), 0, stream>>>(input_image, pc);
  k_knn3<<<dim3(NN / 256, BB), dim3(256), 0, stream>>>(pc, idx);

  // ---------------- stage 1: transform net
  edge_gemm(pc, idx, w64, bigB, 3, 64);            bn(bigB, MEK, 64, g64, b64);
  gemm(bigB, w128, bigA, MEK, 64, 128);            bn(bigA, MEK, 128, g128, b128);
  k_kreduce<<<dim3(gblk((long long)MN * 128)), dim3(256), 0, stream>>>(
      bigA, nmax128, 128, 128, 0, 0, 0);
  gemm(nmax128, w1024, bigA, MN, 128, 1024);       bn(bigA, MN, 1024, g1024, b1024);
  k_nmax<<<dim3(gblk((long long)BB * 1024)), dim3(256), 0, stream>>>(bigA, gfeat, 1024);
  k_fc<<<dim3(gblk((long long)BB * 512)), dim3(256), 0, stream>>>(gfeat, fc1_w, fc1_b, fc1o, BB, 1024, 512);
  k_fc<<<dim3(gblk((long long)BB * 256)), dim3(256), 0, stream>>>(fc1o, fc2_w, fc2_b, fc2o, BB, 512, 256);
  k_tbuild<<<dim3(1), dim3(64), 0, stream>>>(fc2o, tw, tb, T);
  k_tapply<<<dim3(gblk(MN)), dim3(256), 0, stream>>>(pc, T, pct);

  // ---------------- stage 2: feature net on transformed points
  k_knn3<<<dim3(NN / 256, BB), dim3(256), 0, stream>>>(pct, idx);
  edge_gemm(pct, idx, wo1, bigB, 3, 64);           bn(bigB, MEK, 64, go1, bo1);
  gemm(bigB, wo2, bigA, MEK, 64, 64);              bn(bigA, MEK, 64, go2, bo2);
  k_kreduce<<<dim3(gblk((long long)MN * 64)), dim3(256), 0, stream>>>(
      bigA, cat1, 64, 128, 0, 64, 1);
  gemm(cat1, wo3, out3, MN, 128, 64);              bn(out3, MN, 64, go3, bo3);

  // ---------------- stage 3: graph loop (iteration-invariant -> compute once)
  k_concat_if<<<dim3(gblk((long long)MN * 67)), dim3(256), 0, stream>>>(pct, out3, IF);
  gemm(IF, w1, trans1, MN, 67, 3);                 bn(trans1, MN, 3, g1, b1);
  k_knn3<<<dim3(NN / 256, BB), dim3(256), 0, stream>>>(trans1, idx);
  edge_gemm(IF, idx, w2, bigB, 67, 16);            bn(bigB, MEK, 16, g2, b2);
  k_kreduce<<<dim3(gblk((long long)MN * 16)), dim3(256), 0, stream>>>(
      bigB, o5, 16, 32, 0, 16, 1);
  gemm(o5, w3, out5, MN, 32, 16);                  bn(out5, MN, 16, g3, b3);

  // ---------------- pack (replicates the 4 identical graph iterations)
  k_pack<<<dim3(gblk((long long)MN * 16)), dim3(256), 0, stream>>>(out5, o5, out);
}